// GaussianModel_84782654423620
// MI455X (gfx1250) — compile-verified
//
#include <hip/hip_runtime.h>
#include <hip/hip_bf16.h>
#include <math.h>

// ---------------------------------------------------------------------------
// GaussianModel transient histogram for MI455X (gfx1250, wave32).
//   out[b] = ( sum_i clip(e_ib*(P_i + Q_i*d_ib), 0, I_i) ) / r_b^2
// with per-point folded constants:
//   d_ib = r_b - r0_i,  e_ib = exp(k_i * d^2),  k_i = -0.5/sigma_i^2
//   P_i = I_i*cf_i*sqrt(0.5/pi)/sigma_i * h,  Q_i = I_i*(1-cf_i)/sigma_i^2 * h
// Stage 1: 128 blocks x 512 threads, bin-per-lane, LDS-staged point constants.
// Stage 2: deterministic 128-way reduction via V_WMMA_F32_16X16X4_F32.
// ---------------------------------------------------------------------------

#define NBINS 512
#define CHUNK 512
#define NBLK  128
#define HBIN  0.005f            // BIN_RES / 2  (T0 == 0)
#define SQRT_HALF_OVER_PI 0.3989422804014327f

typedef __attribute__((ext_vector_type(2))) float v2f;
typedef __attribute__((ext_vector_type(8))) float v8f;

__global__ __launch_bounds__(CHUNK) void gm_accumulate(
    const float* __restrict__ means,         // [N,3]
    const float* __restrict__ scan_point,    // [3]
    const float* __restrict__ colours,       // [N,1]
    const float* __restrict__ coefficients,  // [N,1]
    const float* __restrict__ opacities,     // [N,1] (view_id==0)
    const float* __restrict__ scales,        // [N,1]
    float* __restrict__ partials,            // [NBLK, NBINS]
    int npoints)
{
    __shared__ float4 pdat[CHUNK];   // {r0, k, P, Q}
    __shared__ float  pint[CHUNK];   // I (clamp ceiling)

    const int   tid = threadIdx.x;            // 0..511 == bin id
    const float rb  = HBIN * (float)(tid + 1);

    const float sx = scan_point[0];
    const float sy = scan_point[1];
    const float sz = scan_point[2];

    float acc = 0.0f;

    for (int base = blockIdx.x * CHUNK; base < npoints; base += gridDim.x * CHUNK) {
        const int i = base + tid;
        __syncthreads();                       // protect LDS reuse across chunks
        if (i < npoints) {
            const float dx = means[3 * i + 0] - sx;
            const float dy = means[3 * i + 1] - sy;
            const float dz = means[3 * i + 2] - sz;
            const float r0 = sqrtf(dx * dx + dy * dy + dz * dz);

            const float sig = fmaxf(__expf(scales[i]), HBIN);
            const float is  = 1.0f / sig;
            const float cf  = 1.0f / (1.0f + __expf(-coefficients[i]));
            const float op  = opacities[i];
            const float co  = colours[i];
            const float I   = (op * op) * (co * co);

            const float P = I * cf * SQRT_HALF_OVER_PI * is * HBIN;
            const float Q = I * (1.0f - cf) * is * is * HBIN;

            pdat[tid] = make_float4(r0, -0.5f * is * is, P, Q);
            pint[tid] = I;
        } else {
            // padding: P=Q=0 and ceiling 0 -> contributes exactly 0
            pdat[tid] = make_float4(0.0f, 0.0f, 0.0f, 0.0f);
            pint[tid] = 0.0f;
        }
        __syncthreads();

        #pragma unroll 4
        for (int j = 0; j < CHUNK; ++j) {       // LDS broadcast reads
            const float4 p = pdat[j];
            const float  d = rb - p.x;
            const float  e = __expf(p.y * (d * d));
            float v = e * fmaf(p.w, d, p.z);
            v = fminf(fmaxf(v, 0.0f), pint[j]);
            acc += v;
        }
    }

    // fixed per-lane summation order -> deterministic
    partials[blockIdx.x * NBINS + tid] = acc;
}

// Reduce 128 block-partials per bin with f32 WMMA:
//   D(16x16) = A(16x4, all ones) x B(4x16 tile of partials) + C
// One wave per 16 bins; 32 accumulating WMMAs sweep all 128 partial rows.
__global__ __launch_bounds__(512) void gm_finalize(
    const float* __restrict__ partials,       // [NBLK, NBINS]
    float* __restrict__ out)                  // [NBINS]
{
    const int tid  = threadIdx.x;             // 0..511
    const int lane = tid & 31;
    const int wave = tid >> 5;                // 0..15
    const int bin0 = wave * 16;

    const int n   = lane & 15;                // column (bin within group)
    const int khi = (lane >> 4) << 1;         // 0 (lanes 0-15) or 2 (lanes 16-31)

    v2f a; a.x = 1.0f; a.y = 1.0f;            // ones: D[m][n] = sum_k B[k][n]
    v8f c = {};

    for (int k0 = 0; k0 < NBLK; k0 += 4) {
        v2f b;
        // B 4x16 layout: VGPR0 = K{0,2}, VGPR1 = K{1,3}, N striped on lanes
        b.x = partials[(k0 + khi + 0) * NBINS + bin0 + n];
        b.y = partials[(k0 + khi + 1) * NBINS + bin0 + n];
        c = __builtin_amdgcn_wmma_f32_16x16x4_f32(
                /*neg_a=*/false, a, /*neg_b=*/false, b,
                /*c_mod=*/(short)0, c, /*reuse_a=*/false, /*reuse_b=*/false);
    }

    // D row M=0 lives in VGPR0 on lanes 0..15 (N = lane)
    if (lane < 16) {
        const int   bin = bin0 + n;
        const float r   = HBIN * (float)(bin + 1);
        out[bin] = c[0] / (r * r);            // DECAY == 2.0
    }
}

extern "C" void kernel_launch(void* const* d_in, const int* in_sizes, int n_in,
                              void* d_out, int out_size, void* d_ws, size_t ws_size,
                              hipStream_t stream)
{
    const float* means  = (const float*)d_in[0];
    const float* scan   = (const float*)d_in[1];
    const float* cols   = (const float*)d_in[2];
    const float* coefs  = (const float*)d_in[3];
    const float* opac   = (const float*)d_in[4];
    const float* scal   = (const float*)d_in[5];
    // d_in[6] = view_id (always 0, VIEW_NUM == 1) -> ignored

    float* out      = (float*)d_out;
    float* partials = (float*)d_ws;           // NBLK*NBINS*4 = 256 KB scratch
    const int npoints = in_sizes[0] / 3;

    gm_accumulate<<<NBLK, CHUNK, 0, stream>>>(means, scan, cols, coefs, opac,
                                              scal, partials, npoints);
    gm_finalize<<<1, 512, 0, stream>>>(partials, out);
}